// CrossCompressUnit_51041391345940
// MI455X (gfx1250) — compile-verified
//
#include <hip/hip_runtime.h>

// CDNA5 / gfx1250 (MI455X), wave32.
// Memory-bound kernel (AI ~ 1 FLOP/byte): single pass, 128-bit accesses,
// non-temporal stores. Dot-product reductions done with V_WMMA_F32_16X16X4_F32
// (two skinny GEMMs: V@[w_ev|w_ee], E@[w_vv|w_ve]), 16 rows per wave.

typedef float v2f __attribute__((ext_vector_type(2)));
typedef float v4f __attribute__((ext_vector_type(4)));
typedef float v8f __attribute__((ext_vector_type(8)));

#define DIM 256
#define ROWS_PER_WAVE 16
#define WAVES_PER_BLOCK 8
#define ROWS_PER_BLOCK (ROWS_PER_WAVE * WAVES_PER_BLOCK)  // 128

__global__ __launch_bounds__(256) void cross_compress_kernel(
    const float* __restrict__ v,  const float* __restrict__ e,
    const float* __restrict__ w_vv, const float* __restrict__ b_vv,
    const float* __restrict__ w_ev, const float* __restrict__ b_ev,
    const float* __restrict__ w_ve, const float* __restrict__ b_ve,
    const float* __restrict__ w_ee, const float* __restrict__ b_ee,
    float* __restrict__ v_out, float* __restrict__ e_out)
{
    const int lane  = threadIdx.x & 31;
    const int wave  = threadIdx.x >> 5;
    const int row0  = (blockIdx.x * WAVES_PER_BLOCK + wave) * ROWS_PER_WAVE;

    // ---------------- Phase 1: WMMA reduction (4 dots per row) --------------
    // A-matrix 16x4 f32 layout: lane L in [0,16) -> M=L, K = {0,1};
    //                           lane L+16       -> M=L, K = {2,3}.
    const int m     = lane & 15;            // A row within tile / B column N
    const int khalf = (lane >> 4) * 2;      // K sub-offset for this lane half

    const float* aV = v + (size_t)(row0 + m) * DIM + khalf;
    const float* aE = e + (size_t)(row0 + m) * DIM + khalf;

    // B-matrix 4x16: column 0 carries one weight vector, column 1 the other,
    // columns 2..15 are zeroed via a per-lane mask (v_cndmask, no EXEC change
    // -> WMMA's EXEC-all-ones requirement is preserved).
    const float* bpV = (m == 1) ? w_ee : w_ev;   // accV: n0 = v.w_ev, n1 = v.w_ee
    const float* bpE = (m == 1) ? w_ve : w_vv;   // accE: n0 = e.w_vv, n1 = e.w_ve
    const float  msk = (m < 2) ? 1.0f : 0.0f;

    v8f accV = {};
    v8f accE = {};

#pragma unroll 8
    for (int k = 0; k < DIM; k += 4) {
        v2f av = *(const v2f*)(aV + k);
        v2f ae = *(const v2f*)(aE + k);
        v2f bv = *(const v2f*)(bpV + khalf + k);
        v2f be = *(const v2f*)(bpE + khalf + k);
        bv.x *= msk; bv.y *= msk;
        be.x *= msk; be.y *= msk;
        // D = A * B + C  (f32 16x16x4)
        accV = __builtin_amdgcn_wmma_f32_16x16x4_f32(
                   false, av, false, bv, (short)0, accV, false, false);
        accE = __builtin_amdgcn_wmma_f32_16x16x4_f32(
                   false, ae, false, be, (short)0, accE, false, false);
    }

    const float c1 = b_vv[0] + b_ev[0];
    const float c2 = b_ve[0] + b_ee[0];

    // ---------------- Phase 2: streaming outputs ----------------------------
    // C/D layout: v8f element r at lane n       = D[M=r  ][N=n]
    //             v8f element r at lane 16+n    = D[M=r+8][N=n]
    // Per-row scalars live in lanes {0,1,16,17}; broadcast with shfl.
#pragma unroll
    for (int r = 0; r < ROWS_PER_WAVE; ++r) {
        const int src = (r < 8) ? 0 : 16;
        const int idx = (r < 8) ? r : (r - 8);
        const float s_ev = __shfl(accV[idx], src + 0, 32);  // v . w_ev
        const float s_ee = __shfl(accV[idx], src + 1, 32);  // v . w_ee
        const float s_vv = __shfl(accE[idx], src + 0, 32);  // e . w_vv
        const float s_ve = __shfl(accE[idx], src + 1, 32);  // e . w_ve

        const size_t base = (size_t)(row0 + r) * DIM;
#pragma unroll
        for (int c = 0; c < 2; ++c) {
            const size_t off = base + (size_t)c * 128 + (size_t)lane * 4;
            v4f vv = *(const v4f*)(v + off);   // hot in WGP$/L2 from phase 1
            v4f ee = *(const v4f*)(e + off);
            v4f vo = vv * s_vv + ee * s_ev + c1;
            v4f eo = vv * s_ve + ee * s_ee + c2;
            // Outputs are never re-read: non-temporal 128-bit stores.
            __builtin_nontemporal_store(vo, (v4f*)(v_out + off));
            __builtin_nontemporal_store(eo, (v4f*)(e_out + off));
        }
    }
}

extern "C" void kernel_launch(void* const* d_in, const int* in_sizes, int n_in,
                              void* d_out, int out_size, void* d_ws, size_t ws_size,
                              hipStream_t stream) {
    (void)n_in; (void)out_size; (void)d_ws; (void)ws_size;

    const float* v    = (const float*)d_in[0];
    const float* e    = (const float*)d_in[1];
    const float* w_vv = (const float*)d_in[2];
    const float* b_vv = (const float*)d_in[3];
    const float* w_ev = (const float*)d_in[4];
    const float* b_ev = (const float*)d_in[5];
    const float* w_ve = (const float*)d_in[6];
    const float* b_ve = (const float*)d_in[7];
    const float* w_ee = (const float*)d_in[8];
    const float* b_ee = (const float*)d_in[9];

    const int rows = in_sizes[0] / DIM;          // B = 131072
    float* vout = (float*)d_out;                 // [B, DIM]
    float* eout = vout + (size_t)rows * DIM;     // [B, DIM]

    dim3 block(256);                              // 8 wave32 waves
    dim3 grid(rows / ROWS_PER_BLOCK);             // 1024 blocks

    hipLaunchKernelGGL(cross_compress_kernel, grid, block, 0, stream,
                       v, e, w_vv, b_vv, w_ev, b_ev, w_ve, b_ve, w_ee, b_ee,
                       vout, eout);
}